// WGCNLayer_64854006169653
// MI455X (gfx1250) — compile-verified
//
#include <hip/hip_runtime.h>
#include <hip/hip_bf16.h>

typedef __attribute__((ext_vector_type(2))) float v2f;
typedef __attribute__((ext_vector_type(8))) float v8f;

#define N_NODES 100000
#define D 128            // D_IN == D_OUT == 128
#define N_EDGES 500000

// ---------------------------------------------------------------------------
// Kernel 1: h = X @ W^T + b, written to both ws (stable copy for edge pass)
// and out (initial value; edge pass accumulates on top).
// One wave per 16x16 output tile, V_WMMA_F32_16X16X4_F32, K-loop 128/4 = 32.
// Tile id t: m0 = (t/8)*16, n0 = (t%8)*16  -> 8 waves of a block share A rows.
// ---------------------------------------------------------------------------
__global__ __launch_bounds__(256) void wgcn_gemm_kernel(
    const float* __restrict__ X,     // N_NODES x 128
    const float* __restrict__ W,     // 128 x 128
    const float* __restrict__ bias,  // 128
    float* __restrict__ h,           // ws: N_NODES x 128
    float* __restrict__ out)         // N_NODES x 128
{
    const int lane = threadIdx.x & 31;
    const int wave = threadIdx.x >> 5;
    const int tile = blockIdx.x * 8 + wave;       // 50000 tiles total
    const int m0   = (tile >> 3) << 4;            // row-tile * 16
    const int n0   = (tile & 7) << 4;             // col-tile * 16
    const int half = lane >> 4;                   // 0: K pair {0,1}, 1: {2,3}
    const int l    = lane & 15;

    // A-matrix per-lane base: row m0+l, starting K offset 2*half
    const float* arow = X + (size_t)(m0 + l) * D + 2 * half;
    // B-matrix per-lane base: B[k][n] = W[n][k] -> row n0+l of W, K offset 2*half
    const float* wrow = W + (size_t)(n0 + l) * D + 2 * half;

    v8f c = {};
#pragma unroll
    for (int k0 = 0; k0 < D; k0 += 4) {
        v2f a  = *(const v2f*)(arow + k0);  // {A[m][k0+2h], A[m][k0+2h+1]}
        v2f bm = *(const v2f*)(wrow + k0);  // {W[n][k0+2h], W[n][k0+2h+1]}
        // 8 args: (neg_a, A, neg_b, B, c_mod, C, reuse_a, reuse_b)
        c = __builtin_amdgcn_wmma_f32_16x16x4_f32(
                false, a, false, bm, (short)0, c, false, false);
    }

    const float bv = bias[n0 + l];
#pragma unroll
    for (int r = 0; r < 8; ++r) {
        // D layout: VGPR r holds rows {m0+r, m0+r+8}, col n0+l
        const int row = m0 + r + 8 * half;
        const float val = c[r] + bv;
        h[(size_t)row * D + n0 + l]   = val;
        out[(size_t)row * D + n0 + l] = val;
    }
}

// ---------------------------------------------------------------------------
// Kernel 2: per edge e: out[dst] += rw * h[src] + (1 - rw) * b
//   (== rw*(X[src]@W^T) + b, the exact per-edge message of the reference)
// One wave per edge; lane handles 4 contiguous floats (float4 gather from h,
// 4x global_atomic_add_f32 scatter into out, which lives in L2).
// ---------------------------------------------------------------------------
__global__ __launch_bounds__(256) void wgcn_edge_kernel(
    const float* __restrict__ h,      // ws: N_NODES x 128 (stable)
    const int*   __restrict__ edges,  // N_EDGES x 3  (src, rel, dst)
    const float* __restrict__ bias,   // 128
    const float* __restrict__ relw,   // 200
    float* __restrict__ out)
{
    const int e    = (int)((blockIdx.x * blockDim.x + threadIdx.x) >> 5);
    const int lane = threadIdx.x & 31;
    if (e >= N_EDGES) return;

    const int src = edges[e * 3 + 0];
    const int rel = edges[e * 3 + 1];
    const int dst = edges[e * 3 + 2];
    const float rw  = relw[rel];
    const float orw = 1.0f - rw;

    const int col = lane * 4;
    const float4 hv = *(const float4*)(h + (size_t)src * D + col);
    const float4 bb = *(const float4*)(bias + col);
    float* o = out + (size_t)dst * D + col;

    unsafeAtomicAdd(o + 0, rw * hv.x + orw * bb.x);
    unsafeAtomicAdd(o + 1, rw * hv.y + orw * bb.y);
    unsafeAtomicAdd(o + 2, rw * hv.z + orw * bb.z);
    unsafeAtomicAdd(o + 3, rw * hv.w + orw * bb.w);
}

extern "C" void kernel_launch(void* const* d_in, const int* in_sizes, int n_in,
                              void* d_out, int out_size, void* d_ws, size_t ws_size,
                              hipStream_t stream) {
    const float* X    = (const float*)d_in[0];  // nodes_embed (100000,128)
    const int*   edges= (const int*)  d_in[1];  // (500000,3)
    const float* W    = (const float*)d_in[2];  // (128,128)
    const float* bias = (const float*)d_in[3];  // (128,)
    const float* relw = (const float*)d_in[4];  // (200,1)
    float* out = (float*)d_out;                 // (100000,128)
    float* h   = (float*)d_ws;                  // 51.2 MB scratch

    // 50000 16x16 tiles, 8 waves (256 thr) per block -> 6250 blocks
    wgcn_gemm_kernel<<<(N_NODES / 16) * (D / 16) / 8, 256, 0, stream>>>(
        X, W, bias, h, out);

    // 500000 edges, one wave each, 8 waves per block -> 62500 blocks
    wgcn_edge_kernel<<<N_EDGES / 8, 256, 0, stream>>>(
        h, edges, bias, relw, out);
}